// DDSPSynth_16466904613675
// MI455X (gfx1250) — compile-verified
//
#include <hip/hip_runtime.h>
#include <math.h>

// ---------------------------------------------------------------------------
// DDSP synth for MI455X (gfx1250, wave32).
//   out layout (floats): [0)        x+n   : 131072
//                        [131072)   tanh  :  65536
//                        [196608)   relu  :  65536
//                        [262144)   x     : 131072
//                        [393216)   n     : 131072
// Workspace (floats): freq_ws 65536 | phase_base 65536 | x_ws 131072 | wav_ws 262144
// ---------------------------------------------------------------------------

#define N_SAMPLES 16384
#define N_OSC     128
#define NB_FRAMES 1024
#define WIN       32
#define HOP       16
#define MAGS      17
#define SR        22050.0f
#define TWO_PI_F  6.283185307179586f
#define INV_2PI_F 0.15915494309189535f

typedef float v2f __attribute__((ext_vector_type(2)));
typedef float v8f __attribute__((ext_vector_type(8)));

__device__ __forceinline__ float relu_f(float x) { return x > 0.0f ? x : 0.0f; }
__device__ __forceinline__ float wrap2pi(float p) {
  return p - TWO_PI_F * floorf(p * INV_2PI_F);
}

// cos(2*pi*m/32) / sqrt(32), m = 0..31  (exact DFT basis, no runtime trig)
__device__ __constant__ float CTAB32[32] = {
    0.17677669529663689f,  0.17337998066526852f,  0.16332037060954706f,
    0.14698445030241986f,  0.12500000000000000f,  0.09821186979300366f,
    0.06764951251827464f,  0.03448742239762385f,  0.0f,
   -0.03448742239762385f, -0.06764951251827464f, -0.09821186979300366f,
   -0.12500000000000000f, -0.14698445030241986f, -0.16332037060954706f,
   -0.17337998066526852f, -0.17677669529663689f, -0.17337998066526852f,
   -0.16332037060954706f, -0.14698445030241986f, -0.12500000000000000f,
   -0.09821186979300366f, -0.06764951251827464f, -0.03448742239762385f,
    0.0f,                  0.03448742239762385f,  0.06764951251827464f,
    0.09821186979300366f,  0.12500000000000000f,  0.14698445030241986f,
    0.16332037060954706f,  0.17337998066526852f};

// ---------------------------------------------------------------------------
// Kernel 1: tanh / relu on params, build per-osc frequency frames.
// ---------------------------------------------------------------------------
__global__ void prep_kernel(const float* __restrict__ freq_params,
                            const float* __restrict__ amp_params,
                            const float* __restrict__ centers,
                            const float* __restrict__ bandwidths,
                            float* __restrict__ out_orig_freq,
                            float* __restrict__ out_orig_amp,
                            float* __restrict__ freq_ws) {
  int i = blockIdx.x * blockDim.x + threadIdx.x;  // 0..65535
  int osc = (i >> 6) & (N_OSC - 1);
  float tf = tanhf(freq_params[i]);
  out_orig_freq[i] = tf;
  freq_ws[i] = centers[osc] + tf * (bandwidths[osc] * 0.5f);
  out_orig_amp[i] = relu_f(amp_params[i]);
}

// ---------------------------------------------------------------------------
// Oscillator pass 1: per (b,osc), wrapped exclusive phase prefix at each
// 256-sample chunk boundary (wave32 shfl reduction -> deterministic).
// ---------------------------------------------------------------------------
__global__ void osc_pass1(const float* __restrict__ freq_ws,
                          float* __restrict__ phase_base) {
  __shared__ float fr[64];
  __shared__ float wsum[8];
  __shared__ float csum[64];
  int bo = blockIdx.x;  // b*128 + osc
  int tid = threadIdx.x;
  int wid = tid >> 5;
  if (tid < 64) fr[tid] = freq_ws[bo * 64 + tid];
  __syncthreads();
  for (int chunk = 0; chunk < 64; ++chunk) {
    int t = chunk * 256 + tid;
    float coords = (t + 0.5f) * (1.0f / 256.0f) - 0.5f;
    coords = fminf(fmaxf(coords, 0.0f), 63.0f);
    int lo = (int)coords;
    int hi = min(lo + 1, 63);
    float w = coords - (float)lo;
    float f = fr[lo] * (1.0f - w) + fr[hi] * w;
    f = fminf(fmaxf(f, 1.0f), SR * 0.5f);
    float v = f * (TWO_PI_F / SR);
#pragma unroll
    for (int off = 16; off > 0; off >>= 1) v += __shfl_xor(v, off, 32);
    if ((tid & 31) == 0) wsum[wid] = v;
    __syncthreads();
    if (tid == 0) {
      float s = 0.0f;
      for (int i = 0; i < 8; ++i) s += wsum[i];
      csum[chunk] = s;
    }
    __syncthreads();
  }
  if (tid == 0) {
    float run = 0.0f;
    for (int c = 0; c < 64; ++c) {
      phase_base[bo * 64 + c] = run;
      run = wrap2pi(run + csum[c]);
    }
  }
}

// ---------------------------------------------------------------------------
// Oscillator pass 2: one block per (b, chunk); fixed-order oscillator loop,
// wave32 shfl inclusive scan + cross-wave LDS prefix; wrapped phase + native
// cos. Deterministic (no atomics).
// ---------------------------------------------------------------------------
__global__ void osc_pass2(const float* __restrict__ freq_ws,
                          const float* __restrict__ amp_params,
                          const float* __restrict__ phase_base,
                          float* __restrict__ x_ws) {
  __shared__ float fr[64];
  __shared__ float am[64];
  __shared__ float wsum[8];
  int b = blockIdx.x >> 6;
  int chunk = blockIdx.x & 63;
  int tid = threadIdx.x;
  int wid = tid >> 5;
  int t = chunk * 256 + tid;
  float coords = (t + 0.5f) * (1.0f / 256.0f) - 0.5f;
  coords = fminf(fmaxf(coords, 0.0f), 63.0f);
  int lo = (int)coords;
  int hi = min(lo + 1, 63);
  float w = coords - (float)lo;
  float acc = 0.0f;
  for (int osc = 0; osc < N_OSC; ++osc) {
    int base = (b * N_OSC + osc) * 64;
    __syncthreads();  // fr/am/wsum free from previous oscillator
    if (tid < 64) {
      fr[tid] = freq_ws[base + tid];
      am[tid] = relu_f(amp_params[base + tid]);
    }
    __syncthreads();
    float f = fr[lo] * (1.0f - w) + fr[hi] * w;
    f = fminf(fmaxf(f, 1.0f), SR * 0.5f);
    float v = f * (TWO_PI_F / SR);
#pragma unroll
    for (int off = 1; off < 32; off <<= 1) {
      float up = __shfl_up(v, off, 32);
      if ((tid & 31) >= off) v += up;
    }
    if ((tid & 31) == 31) wsum[wid] = v;
    __syncthreads();
    float wpre = 0.0f;
    for (int i = 0; i < wid; ++i) wpre += wsum[i];
    float phase = wrap2pi(phase_base[base + chunk] + wpre + v);
    float a = am[lo] * (1.0f - w) + am[hi] * w;
    acc += __cosf(phase) * a;
  }
  x_ws[b * N_SAMPLES + t] = acc;
}

// ---------------------------------------------------------------------------
// Noise bank via f32 WMMA DFT matmuls. One wave per 16-frame tile.
//  Stage A: [16x32] nf @ [32x32] {cos_k | Re16-basis,-sin_k} -> Re, IX tiles
//  scale by relu(noise_params), repack in LDS as [16x40] (stride 41)
//  Stage B: [16x40] RI @ [40x32] IDFT -> wav tile.
// Frame tile fetched with async global->LDS (ASYNCcnt) b128 loads.
// All B-fragment coefficient selection is branchless (v_cndmask, no EXEC
// churn around v_wmma).
// ---------------------------------------------------------------------------
__device__ __forceinline__ float idft_coef_t(const float* ctab, int kk, int j) {
  // Padded K columns (17..19, 36..39) multiply a guaranteed-zero A operand,
  // so their B value is don't-care; only two real cases remain.
  bool fwd = kk < 17;
  int idx = fwd ? ((kk * j) & 31) : (((kk - 20) * j - 8) & 31);
  float scale = fwd ? (((kk & 15) == 0) ? 1.0f : 2.0f) : -2.0f;
  return scale * ctab[idx];
}

__global__ void noise_wmma_kernel(const float* __restrict__ noise,
                                  const float* __restrict__ noise_params,
                                  float* __restrict__ wav_ws) {
  __shared__ float nf_t[16][36];  // 16 frames x 32 samples, 144B row stride
  __shared__ float ri[16][41];    // [Re'_0..16 | 0 0 0 | Im'_0..15 | 0 0 0 0]
  __shared__ float ctab[32];
  int tile = blockIdx.x;    // 0..511
  int b = tile >> 6;        // 64 tiles per batch (1024 frames)
  int f0 = (tile & 63) * 16;
  int lane = threadIdx.x;   // 0..31
  int nloc = lane & 15;
  int khi = (lane >> 4) * 2;

  ctab[lane] = CTAB32[lane];

  // Async-load frame tile: 4 x b128 (4 floats/lane); zero-fill OOB tail slots.
#pragma unroll
  for (int q = 0; q < 4; ++q) {
    int e = q * 128 + lane * 4;  // element in 512-float tile
    int row = e >> 5;
    int col = e & 31;            // multiple of 4 -> 16B aligned
    int s = (f0 + row) * HOP + col;
    unsigned lds_a = (unsigned)(size_t)&nf_t[row][col];
    if (s < N_SAMPLES) {
      unsigned long long ga = (unsigned long long)(size_t)(noise + b * N_SAMPLES + s);
      asm volatile("global_load_async_to_lds_b128 %0, %1, off"
                   :: "v"(lds_a), "v"(ga) : "memory");
    } else {
      nf_t[row][col] = 0.0f; nf_t[row][col + 1] = 0.0f;
      nf_t[row][col + 2] = 0.0f; nf_t[row][col + 3] = 0.0f;
    }
  }
  asm volatile("s_wait_asynccnt 0x0" ::: "memory");
  __syncthreads();

  // ---- Stage A: forward rDFT (ortho) ----
  v8f cRe = {0, 0, 0, 0, 0, 0, 0, 0};
  v8f cIX = {0, 0, 0, 0, 0, 0, 0, 0};  // col0: Re16 basis; cols>=1: Im_k
  bool is0 = (nloc == 0);
#pragma unroll
  for (int kb = 0; kb < 8; ++kb) {
    int j0 = 4 * kb + khi;
    v2f a;
    a[0] = nf_t[nloc][j0 + 0];
    a[1] = nf_t[nloc][j0 + 1];
    int m0 = (nloc * j0) & 31;
    int m1 = (nloc * (j0 + 1)) & 31;
    v2f b0, b1;
    b0[0] = ctab[m0];
    b0[1] = ctab[m1];
    // tile1: col0 = cos(pi*j)/sqrt32 (Re16 basis, also a table entry);
    //        cols>=1 = -sin(2*pi*k*j/32)/sqrt32 = -ctab[(k*j-8)&31].
    int i0 = is0 ? ((16 * j0) & 31) : ((m0 - 8) & 31);
    int i1 = is0 ? ((16 * (j0 + 1)) & 31) : ((m1 - 8) & 31);
    float t0 = ctab[i0];
    float t1 = ctab[i1];
    b1[0] = is0 ? t0 : -t0;
    b1[1] = is0 ? t1 : -t1;
    cRe = __builtin_amdgcn_wmma_f32_16x16x4_f32(false, a, false, b0, (short)0, cRe, false, false);
    cIX = __builtin_amdgcn_wmma_f32_16x16x4_f32(false, a, false, b1, (short)0, cIX, false, false);
  }
  __syncthreads();

  // ---- Scale by relu(noise_params[b,k,f]), repack into stage-B A layout ----
#pragma unroll
  for (int r = 0; r < 8; ++r) {
    int M = r + ((lane >> 4) << 3);  // row 0..15 per C/D layout
    int fl = f0 + M;
    float cf = relu_f(noise_params[(b * MAGS + nloc) * NB_FRAMES + fl]);
    ri[M][nloc] = cRe[r] * cf;  // Re'_k, k = nloc
    if (nloc == 0) {
      float cf16 = relu_f(noise_params[(b * MAGS + 16) * NB_FRAMES + fl]);
      ri[M][16] = cIX[r] * cf16;  // Re'_16
      ri[M][20] = 0.0f;           // Im'_0 == 0
    } else {
      ri[M][20 + nloc] = cIX[r] * cf;  // Im'_k
    }
    if (nloc >= 1 && nloc <= 3) ri[M][16 + nloc] = 0.0f;  // cols 17..19
    if (nloc >= 12) ri[M][24 + nloc] = 0.0f;              // cols 36..39
  }
  __syncthreads();

  // ---- Stage B: inverse rDFT (ortho), K = 40 padded ----
  v8f w0 = {0, 0, 0, 0, 0, 0, 0, 0};
  v8f w1 = {0, 0, 0, 0, 0, 0, 0, 0};
#pragma unroll
  for (int ks = 0; ks < 10; ++ks) {
    int kk0 = 4 * ks + khi;
    v2f a;
    a[0] = ri[nloc][kk0 + 0];
    a[1] = ri[nloc][kk0 + 1];
    v2f b0, b1;
    b0[0] = idft_coef_t(ctab, kk0, nloc);
    b0[1] = idft_coef_t(ctab, kk0 + 1, nloc);
    b1[0] = idft_coef_t(ctab, kk0, nloc + 16);
    b1[1] = idft_coef_t(ctab, kk0 + 1, nloc + 16);
    w0 = __builtin_amdgcn_wmma_f32_16x16x4_f32(false, a, false, b0, (short)0, w0, false, false);
    w1 = __builtin_amdgcn_wmma_f32_16x16x4_f32(false, a, false, b1, (short)0, w1, false, false);
  }

  // ---- Store wav tile (OLA handled by combine kernel; deterministic) ----
#pragma unroll
  for (int r = 0; r < 8; ++r) {
    int M = r + ((lane >> 4) << 3);
    int row = (b * NB_FRAMES + f0 + M) * WIN;
    wav_ws[row + nloc] = w0[r];
    wav_ws[row + 16 + nloc] = w1[r];
  }
}

// ---------------------------------------------------------------------------
// Final: 50% overlap-add of wav frames + assemble all outputs.
// ---------------------------------------------------------------------------
__global__ void combine_kernel(const float* __restrict__ x_ws,
                               const float* __restrict__ wav_ws,
                               float* __restrict__ out) {
  int i = blockIdx.x * blockDim.x + threadIdx.x;  // 0..131071
  int b = i >> 14;
  int s = i & (N_SAMPLES - 1);
  int f = s >> 4;
  int h = s & 15;
  float nv = wav_ws[(b * NB_FRAMES + f) * WIN + h];
  if (f > 0) nv += wav_ws[(b * NB_FRAMES + f - 1) * WIN + 16 + h];
  float xv = x_ws[i];
  out[i] = xv + nv;      // x + n
  out[262144 + i] = xv;  // x
  out[393216 + i] = nv;  // n
}

// ---------------------------------------------------------------------------
extern "C" void kernel_launch(void* const* d_in, const int* in_sizes, int n_in,
                              void* d_out, int out_size, void* d_ws, size_t ws_size,
                              hipStream_t stream) {
  (void)in_sizes; (void)n_in; (void)out_size; (void)ws_size;
  const float* freq_params = (const float*)d_in[0];
  const float* amp_params = (const float*)d_in[1];
  const float* noise_params = (const float*)d_in[2];
  const float* centers = (const float*)d_in[3];
  const float* bandwidths = (const float*)d_in[4];
  const float* noise = (const float*)d_in[5];
  float* out = (float*)d_out;
  float* ws = (float*)d_ws;
  float* freq_ws = ws;              // 65536
  float* phase_base = ws + 65536;   // 65536
  float* x_ws = ws + 131072;        // 131072
  float* wav_ws = ws + 262144;      // 262144

  prep_kernel<<<256, 256, 0, stream>>>(freq_params, amp_params, centers, bandwidths,
                                       out + 131072, out + 196608, freq_ws);
  osc_pass1<<<1024, 256, 0, stream>>>(freq_ws, phase_base);
  osc_pass2<<<512, 256, 0, stream>>>(freq_ws, amp_params, phase_base, x_ws);
  noise_wmma_kernel<<<512, 32, 0, stream>>>(noise, noise_params, wav_ws);
  combine_kernel<<<512, 256, 0, stream>>>(x_ws, wav_ws, out);
}